// MonotonicAlignment_78400333021280
// MI455X (gfx1250) — compile-verified
//
#include <hip/hip_runtime.h>
#include <math.h>

typedef float v2f __attribute__((ext_vector_type(2)));
typedef float v8f __attribute__((ext_vector_type(8)));

#define B_   32
#define S_   2048
#define D_   1024
#define U_   1024
#define WIN  7      // WINDOW_SIZE/2 * 2 + 1
#define MROWS 256   // 32 query rows + 32*7 window key rows

// -----------------------------------------------------------------------------
// Kernel 1: P[256][1024] = [ query ; gathered window keys ] @ [Wa ; Ua] + bias
//   row tiles 0..1  -> query rows,       weight = Wa_w, bias = Wa_b
//   row tiles 2..15 -> window key rows,  weight = Ua_w, bias = Ua_b
// One wave per 16x16 output tile, K-loop of V_WMMA_F32_16X16X4_F32.
// Branchless masking + 1-iteration software pipeline on fragment loads.
// -----------------------------------------------------------------------------
__global__ __launch_bounds__(128)
void bahdanau_proj_wmma(const float* __restrict__ query,
                        const float* __restrict__ keys,
                        const float* __restrict__ Wa_w,
                        const float* __restrict__ Wa_b,
                        const float* __restrict__ Ua_w,
                        const float* __restrict__ Ua_b,
                        const int*   __restrict__ tptr,
                        float*       __restrict__ P)
{
    const int wave = threadIdx.x >> 5;          // 0..3
    const int lane = threadIdx.x & 31;
    const int m    = lane & 15;                 // M (A) / N (B,C,D) index
    const int half = lane >> 4;                 // 0 or 1

    const int rowTile = blockIdx.y;             // 0..15
    const int colTile = blockIdx.x * 4 + wave;  // 0..63
    const int n = colTile * 16 + m;             // output column (0..1023)

    const int t = tptr[0];
    int start = t - (WIN / 2); if (start < 0)  start = 0;
    int end   = t + (WIN / 2) + 1; if (end > S_) end = S_;

    const float* W;
    const float* bias;
    const float* arow;
    float amask = 1.0f;                         // 0 -> masked (zero) A row
    if (rowTile < 2) {
        W = Wa_w; bias = Wa_b;
        arow = query + (size_t)(rowTile * 16 + m) * D_;
    } else {
        W = Ua_w; bias = Ua_b;
        const int g = rowTile * 16 + m - 32;    // 0..223
        const int b = g / WIN;
        const int w = g % WIN;
        const int s = start + w;
        const int sc = (s < S_) ? s : (S_ - 1); // always-safe address
        if (s >= end) amask = 0.0f;
        arow = keys + ((size_t)b * S_ + (size_t)sc) * D_;
    }
    arow += half * 2;                           // this lane's K sub-offset

    const float* bcol = W + n + (size_t)(half * 2) * U_;   // W[k][n] at k*U_+n

    v8f acc = {};

    // A frag: lane holds M = m; VGPR0/1 hold K = kk+2*half {+0,+1}  (contiguous)
    // B frag: lane holds N = n; VGPR0/1 hold K = kk+2*half {+0,+1}  (stride U_)
    v2f a_cur, b_cur, a_nxt, b_nxt;
    a_cur = *(const v2f*)arow;
    b_cur.x = bcol[0];
    b_cur.y = bcol[U_];

    #pragma unroll 4
    for (int kk = 0; kk < D_; kk += 4) {
        if (kk + 4 < D_) {                      // prefetch next fragments
            a_nxt = *(const v2f*)(arow + (kk + 4));
            b_nxt.x = bcol[(size_t)(kk + 4) * U_];
            b_nxt.y = bcol[(size_t)(kk + 5) * U_];
        }
        v2f a = a_cur * amask;                  // branchless window mask
        acc = __builtin_amdgcn_wmma_f32_16x16x4_f32(
                  /*neg_a=*/false, a, /*neg_b=*/false, b_cur,
                  /*c_mod=*/(short)0, acc, /*reuse_a=*/false, /*reuse_b=*/false);
        a_cur = a_nxt;
        b_cur = b_nxt;
    }

    // C/D layout: VGPR v -> row (v + 8*half), col n
    const float bv = bias[n];
    float* pcol = P + n;
    #pragma unroll
    for (int v = 0; v < 8; ++v) {
        const int row = rowTile * 16 + v + half * 8;
        pcol[(size_t)row * U_] = acc[v] + bv;
    }
}

// -----------------------------------------------------------------------------
// Kernel 2: per-batch scores, softmax (closed form over masked positions),
// attention-weight map, and windowed context sum.
//   wave 0   : score_out = tanh(q_proj + Ua_b) . Va + Va_b  (all masked s)
//   wave 1..7: score_in[w] = tanh(q_proj + k_proj[w]) . Va + Va_b
// -----------------------------------------------------------------------------
__global__ __launch_bounds__(256)
void bahdanau_score_ctx(const float* __restrict__ values,
                        const float* __restrict__ Ua_b,
                        const float* __restrict__ Va_w,
                        const float* __restrict__ Va_b,
                        const int*   __restrict__ tptr,
                        const float* __restrict__ P,
                        float*       __restrict__ out)
{
    __shared__ float s_scores[8];

    const int b    = blockIdx.x;
    const int tid  = threadIdx.x;
    const int wave = tid >> 5;
    const int lane = tid & 31;

    const int t = tptr[0];
    int start = t - (WIN / 2); if (start < 0)  start = 0;
    int end   = t + (WIN / 2) + 1; if (end > S_) end = S_;
    const int L = end - start;

    const float* qrow  = P + (size_t)b * U_;
    const float* other = (wave == 0)
                       ? Ua_b
                       : (P + (size_t)(32 + b * WIN + (wave - 1)) * U_);

    float acc = 0.0f;
    for (int u = lane; u < U_; u += 32)
        acc += tanhf(qrow[u] + other[u]) * Va_w[u];
    #pragma unroll
    for (int off = 16; off > 0; off >>= 1)
        acc += __shfl_xor(acc, off, 32);
    if (lane == 0) s_scores[wave] = acc + Va_b[0];
    __syncthreads();

    const float so = s_scores[0];
    float mx = so;
    for (int w = 0; w < L; ++w) mx = fmaxf(mx, s_scores[1 + w]);

    const float eo = __expf(so - mx);
    float denom = (float)(S_ - L) * eo;
    float ein[WIN];
    #pragma unroll
    for (int w = 0; w < WIN; ++w) {
        const float e = (w < L) ? __expf(s_scores[1 + w] - mx) : 0.0f;
        ein[w] = e;
        denom += e;
    }
    const float inv   = 1.0f / denom;
    const float a_out = eo * inv;

    // attention_weights [B, S, 1] lives after context [B, D] in d_out
    float* attn = out + (size_t)B_ * D_ + (size_t)b * S_;
    for (int s = tid; s < S_; s += 256) {
        float v = a_out;
        const int w = s - start;
        if (w >= 0 && w < L) v = ein[w] * inv;
        attn[s] = v;
    }

    // context [B, D]: only window rows of values are non-zero
    float* ctx = out + (size_t)b * D_;
    for (int d = tid; d < D_; d += 256) {
        float c = 0.0f;
        for (int w = 0; w < L; ++w)
            c += (ein[w] * inv) *
                 values[((size_t)b * S_ + (size_t)(start + w)) * D_ + d];
        ctx[d] = c;
    }
}

extern "C" void kernel_launch(void* const* d_in, const int* in_sizes, int n_in,
                              void* d_out, int out_size, void* d_ws, size_t ws_size,
                              hipStream_t stream)
{
    (void)in_sizes; (void)n_in; (void)out_size; (void)ws_size;

    const float* query  = (const float*)d_in[0];
    const float* keys   = (const float*)d_in[1];
    const float* values = (const float*)d_in[2];
    const float* Wa_w   = (const float*)d_in[3];
    const float* Wa_b   = (const float*)d_in[4];
    const float* Ua_w   = (const float*)d_in[5];
    const float* Ua_b   = (const float*)d_in[6];
    const float* Va_w   = (const float*)d_in[7];
    const float* Va_b   = (const float*)d_in[8];
    const int*   tptr   = (const int*)d_in[9];

    float* out = (float*)d_out;
    float* P   = (float*)d_ws;      // MROWS * U_ floats = 1 MB scratch

    // 64 col tiles (4 waves/block * 16 blocks) x 16 row tiles
    dim3 gridA(16, 16);
    bahdanau_proj_wmma<<<gridA, 128, 0, stream>>>(query, keys, Wa_w, Wa_b,
                                                  Ua_w, Ua_b, tptr, P);
    bahdanau_score_ctx<<<B_, 256, 0, stream>>>(values, Ua_b, Va_w, Va_b,
                                               tptr, P, out);
}